// PhyloGAT_26886495273473
// MI455X (gfx1250) — compile-verified
//
#include <hip/hip_runtime.h>

typedef __attribute__((ext_vector_type(2))) float v2f;
typedef __attribute__((ext_vector_type(8))) float v8f;

#define IN_F   128
#define HIDL   64
#define HEADS  4
#define C1     256   // HEADS*HIDL
#define OUTF   32
#define NEG_SLOPE 0.2f
#define BN_EPS 1e-5f

__device__ __forceinline__ float lrelu(float v) { return v > 0.f ? v : NEG_SLOPE * v; }

// ---------------- CSR build ----------------
__global__ void k_zero_i32(int* p, int n) {
    int i = blockIdx.x * blockDim.x + threadIdx.x;
    if (i < n) p[i] = 0;
}

__global__ void k_count(const int* __restrict__ dst, int* __restrict__ cnt, int E) {
    int i = blockIdx.x * blockDim.x + threadIdx.x;
    if (i < E) atomicAdd(&cnt[dst[i]], 1);
}

// single-block exclusive scan (n up to ~100k is fine)
__global__ __launch_bounds__(1024) void k_scan(const int* __restrict__ cnt,
                                               int* __restrict__ offs, int n) {
    __shared__ int buf[1024];
    __shared__ int running;
    if (threadIdx.x == 0) running = 0;
    __syncthreads();
    for (int base = 0; base < n; base += 1024) {
        int i = base + (int)threadIdx.x;
        int v = (i < n) ? cnt[i] : 0;
        buf[threadIdx.x] = v;
        __syncthreads();
        for (int d = 1; d < 1024; d <<= 1) {
            int t2 = 0;
            if ((int)threadIdx.x >= d) t2 = buf[threadIdx.x - d];
            __syncthreads();
            buf[threadIdx.x] += t2;
            __syncthreads();
        }
        int incl = buf[threadIdx.x];
        if (i < n) offs[i] = running + (incl - v);
        __syncthreads();
        if (threadIdx.x == 1023) running += incl;
        __syncthreads();
    }
    if (threadIdx.x == 0) offs[n] = running;
}

__global__ void k_fill(const int* __restrict__ src, const int* __restrict__ dst,
                       const int* __restrict__ offs, int* __restrict__ cur,
                       int* __restrict__ csr_src, int E) {
    int i = blockIdx.x * blockDim.x + threadIdx.x;
    if (i < E) {
        int d = dst[i];
        int pos = offs[d] + atomicAdd(&cur[d], 1);
        csr_src[pos] = src[i];
    }
}

// ---------------- GEMM1: H = X @ W1 ; fused al_s / al_d ----------------
// grid (N/16, HEADS), block 128 (4 waves); wave w -> 16-col tile inside head
__global__ __launch_bounds__(128) void gat_gemm1(
    const float* __restrict__ X, const float* __restrict__ W,
    const float* __restrict__ a_src, const float* __restrict__ a_dst,
    float* __restrict__ H, float* __restrict__ ALS, float* __restrict__ ALD) {
    const int row0 = blockIdx.x * 16;
    const int head = blockIdx.y;
    const int wave = threadIdx.x >> 5;
    const int lane = threadIdx.x & 31;
    const int ln   = lane & 15;
    const int hi   = lane >> 4;
    const int lcol = wave * 16 + ln;           // column within head (0..63)
    const int gcol = head * HIDL + lcol;       // global column (0..255)

    v8f acc = {};
    const float* xr = X + (size_t)(row0 + ln) * IN_F;
#pragma unroll 4
    for (int k = 0; k < IN_F; k += 4) {
        int ka = k + 2 * hi;
        v2f a, b;
        a.x = xr[ka];
        a.y = xr[ka + 1];
        b.x = W[(size_t)ka * C1 + gcol];
        b.y = W[(size_t)(ka + 1) * C1 + gcol];
        acc = __builtin_amdgcn_wmma_f32_16x16x4_f32(false, a, false, b,
                                                    (short)0, acc, false, false);
    }
    // store H tile: VGPR i holds row (hi*8 + i), col = ln
#pragma unroll
    for (int i = 0; i < 8; ++i)
        H[(size_t)(row0 + hi * 8 + i) * C1 + gcol] = acc[i];

    // attention score partials: al[n][head] = sum_f h[n][head][f]*a[head][f]
    float as = a_src[head * HIDL + lcol];
    float ad = a_dst[head * HIDL + lcol];
    float ps[8], pd[8];
#pragma unroll
    for (int i = 0; i < 8; ++i) { ps[i] = acc[i] * as; pd[i] = acc[i] * ad; }
#pragma unroll
    for (int off = 1; off < 16; off <<= 1) {
#pragma unroll
        for (int i = 0; i < 8; ++i) {
            ps[i] += __shfl_xor(ps[i], off, 32);
            pd[i] += __shfl_xor(pd[i], off, 32);
        }
    }
    __shared__ float s_as[16], s_ad[16];
    if (threadIdx.x < 16) { s_as[threadIdx.x] = 0.f; s_ad[threadIdx.x] = 0.f; }
    __syncthreads();
    if (ln == 0) {
#pragma unroll
        for (int i = 0; i < 8; ++i) {
            atomicAdd(&s_as[hi * 8 + i], ps[i]);
            atomicAdd(&s_ad[hi * 8 + i], pd[i]);
        }
    }
    __syncthreads();
    if (threadIdx.x < 16) {
        ALS[(size_t)(row0 + threadIdx.x) * HEADS + head] = s_as[threadIdx.x];
        ALD[(size_t)(row0 + threadIdx.x) * HEADS + head] = s_ad[threadIdx.x];
    }
}

// ---------------- Layer-1 aggregation + bias + BN + ELU ----------------
// one block (128 thr) per dst node; thread t covers contiguous features 2t, 2t+1
// (both inside head t>>5), so the gather is one coalesced float2 per edge.
__global__ __launch_bounds__(128) void gat_agg1(
    const float* __restrict__ H, const float* __restrict__ ALS, const float* __restrict__ ALD,
    const int* __restrict__ offs, const int* __restrict__ csr_src,
    const float* __restrict__ b1, const float* __restrict__ gam, const float* __restrict__ bet,
    const float* __restrict__ bmean, const float* __restrict__ bvar,
    float* __restrict__ X1) {
    const int node = blockIdx.x;
    const int t = threadIdx.x;
    const int beg = offs[node], end = offs[node + 1];

    __shared__ float s_max[HEADS], s_den[HEADS], s_wself[HEADS];
    __shared__ float s_w[32][HEADS];
    __shared__ int s_src[32];

    if (t < HEADS) {
        int h = t;
        float adn = ALD[(size_t)node * HEADS + h];
        float lself = lrelu(ALS[(size_t)node * HEADS + h] + adn);
        float mx = lself;
        for (int e = beg; e < end; ++e) {
            int s = csr_src[e];
            mx = fmaxf(mx, lrelu(ALS[(size_t)s * HEADS + h] + adn));
        }
        s_max[h] = mx;
        float ws = __expf(lself - mx);
        s_wself[h] = ws;
        s_den[h] = ws;
    }
    __syncthreads();

    const int c0 = 2 * t;          // contiguous feature pair
    const int hd = t >> 5;         // head of both features (c0/64)
    float2 acc = make_float2(0.f, 0.f);

    for (int e0 = beg; e0 < end; e0 += 32) {
        int m = end - e0; if (m > 32) m = 32;
        if (t < m) {
            int s = csr_src[e0 + t];
            s_src[t] = s;
#pragma unroll
            for (int h = 0; h < HEADS; ++h) {
                float w = __expf(lrelu(ALS[(size_t)s * HEADS + h] +
                                       ALD[(size_t)node * HEADS + h]) - s_max[h]);
                s_w[t][h] = w;
                atomicAdd(&s_den[h], w);
            }
        }
        __syncthreads();
        for (int e = 0; e < m; ++e) {
            const float2* hs = (const float2*)(H + (size_t)s_src[e] * C1);
            if (e + 1 < m) __builtin_prefetch(H + (size_t)s_src[e + 1] * C1, 0, 0);
            float w = s_w[e][hd];
            float2 v = hs[t];
            acc.x += w * v.x;
            acc.y += w * v.y;
        }
        __syncthreads();
    }
    { // self loop message
        const float2* hs = (const float2*)(H + (size_t)node * C1);
        float w = s_wself[hd];
        float2 v = hs[t];
        acc.x += w * v.x;
        acc.y += w * v.y;
    }
    // finalize: /denom + b1, BatchNorm (running stats), ELU
    {
        float invden = 1.f / (s_den[hd] + 1e-16f);
        float v = acc.x * invden + b1[c0];
        v = (v - bmean[c0]) * rsqrtf(bvar[c0] + BN_EPS) * gam[c0] + bet[c0];
        X1[(size_t)node * C1 + c0] = v > 0.f ? v : (__expf(v) - 1.f);
        v = acc.y * invden + b1[c0 + 1];
        v = (v - bmean[c0 + 1]) * rsqrtf(bvar[c0 + 1] + BN_EPS) * gam[c0 + 1] + bet[c0 + 1];
        X1[(size_t)node * C1 + c0 + 1] = v > 0.f ? v : (__expf(v) - 1.f);
    }
}

// ---------------- GEMM2: H2 = X1 @ W2 ; fused al2_s / al2_d ----------------
// grid N/16, block 64 (2 waves), each wave one 16-col tile of the 32 outputs
__global__ __launch_bounds__(64) void gat_gemm2(
    const float* __restrict__ X1, const float* __restrict__ W,
    const float* __restrict__ a_src, const float* __restrict__ a_dst,
    float* __restrict__ H2, float* __restrict__ ALS2, float* __restrict__ ALD2) {
    const int row0 = blockIdx.x * 16;
    const int wave = threadIdx.x >> 5;
    const int lane = threadIdx.x & 31;
    const int ln   = lane & 15;
    const int hi   = lane >> 4;
    const int gcol = wave * 16 + ln;

    v8f acc = {};
    const float* xr = X1 + (size_t)(row0 + ln) * C1;
#pragma unroll 4
    for (int k = 0; k < C1; k += 4) {
        int ka = k + 2 * hi;
        v2f a, b;
        a.x = xr[ka];
        a.y = xr[ka + 1];
        b.x = W[(size_t)ka * OUTF + gcol];
        b.y = W[(size_t)(ka + 1) * OUTF + gcol];
        acc = __builtin_amdgcn_wmma_f32_16x16x4_f32(false, a, false, b,
                                                    (short)0, acc, false, false);
    }
#pragma unroll
    for (int i = 0; i < 8; ++i)
        H2[(size_t)(row0 + hi * 8 + i) * OUTF + gcol] = acc[i];

    float as = a_src[gcol];
    float ad = a_dst[gcol];
    float ps[8], pd[8];
#pragma unroll
    for (int i = 0; i < 8; ++i) { ps[i] = acc[i] * as; pd[i] = acc[i] * ad; }
#pragma unroll
    for (int off = 1; off < 16; off <<= 1) {
#pragma unroll
        for (int i = 0; i < 8; ++i) {
            ps[i] += __shfl_xor(ps[i], off, 32);
            pd[i] += __shfl_xor(pd[i], off, 32);
        }
    }
    __shared__ float s_as[16], s_ad[16];
    if (threadIdx.x < 16) { s_as[threadIdx.x] = 0.f; s_ad[threadIdx.x] = 0.f; }
    __syncthreads();
    if (ln == 0) {
#pragma unroll
        for (int i = 0; i < 8; ++i) {
            atomicAdd(&s_as[hi * 8 + i], ps[i]);
            atomicAdd(&s_ad[hi * 8 + i], pd[i]);
        }
    }
    __syncthreads();
    if (threadIdx.x < 16) {
        ALS2[row0 + threadIdx.x] = s_as[threadIdx.x];
        ALD2[row0 + threadIdx.x] = s_ad[threadIdx.x];
    }
}

// ---------------- Layer-2 aggregation (1 head, 32 feats) ----------------
__global__ __launch_bounds__(32) void gat_agg2(
    const float* __restrict__ H2, const float* __restrict__ ALS2, const float* __restrict__ ALD2,
    const int* __restrict__ offs, const int* __restrict__ csr_src,
    const float* __restrict__ b2, float* __restrict__ OUT) {
    const int node = blockIdx.x;
    const int t = threadIdx.x;
    const int beg = offs[node], end = offs[node + 1];
    __shared__ float s_max, s_den, s_wself;
    __shared__ float s_w[32];
    __shared__ int s_src[32];

    float adn = ALD2[node];
    if (t == 0) {
        float lself = lrelu(ALS2[node] + adn);
        float mx = lself;
        for (int e = beg; e < end; ++e)
            mx = fmaxf(mx, lrelu(ALS2[csr_src[e]] + adn));
        s_max = mx;
        float ws = __expf(lself - mx);
        s_wself = ws;
        s_den = ws;
    }
    __syncthreads();

    float acc = 0.f;
    for (int e0 = beg; e0 < end; e0 += 32) {
        int m = end - e0; if (m > 32) m = 32;
        if (t < m) {
            int s = csr_src[e0 + t];
            s_src[t] = s;
            float w = __expf(lrelu(ALS2[s] + adn) - s_max);
            s_w[t] = w;
            atomicAdd(&s_den, w);
        }
        __syncthreads();
        for (int e = 0; e < m; ++e)
            acc += s_w[e] * H2[(size_t)s_src[e] * OUTF + t];
        __syncthreads();
    }
    acc += s_wself * H2[(size_t)node * OUTF + t];
    OUT[(size_t)node * OUTF + t] = acc / (s_den + 1e-16f) + b2[t];
}

// ---------------- launcher ----------------
extern "C" void kernel_launch(void* const* d_in, const int* in_sizes, int n_in,
                              void* d_out, int out_size, void* d_ws, size_t ws_size,
                              hipStream_t stream) {
    const float* x      = (const float*)d_in[0];
    const int*   ei     = (const int*)d_in[1];
    const float* W1     = (const float*)d_in[2];
    const float* a_src1 = (const float*)d_in[3];
    const float* a_dst1 = (const float*)d_in[4];
    const float* b1     = (const float*)d_in[5];
    const float* gam    = (const float*)d_in[6];
    const float* bet    = (const float*)d_in[7];
    const float* bmean  = (const float*)d_in[8];
    const float* bvar   = (const float*)d_in[9];
    const float* W2     = (const float*)d_in[10];
    const float* a_src2 = (const float*)d_in[11];
    const float* a_dst2 = (const float*)d_in[12];
    const float* b2     = (const float*)d_in[13];
    float* out = (float*)d_out;

    const int N = in_sizes[0] / IN_F;   // 50000
    const int E = in_sizes[1] / 2;      // 800000
    const int* e_src = ei;
    const int* e_dst = ei + E;

    // workspace layout
    float* h1   = (float*)d_ws;
    float* x1   = h1 + (size_t)N * C1;
    float* h2   = x1 + (size_t)N * C1;
    float* als1 = h2 + (size_t)N * OUTF;
    float* ald1 = als1 + (size_t)N * HEADS;
    float* als2 = ald1 + (size_t)N * HEADS;
    float* ald2 = als2 + N;
    int* cnt  = (int*)(ald2 + N);
    int* offs = cnt + N;
    int* cur  = offs + (N + 1);
    int* csr  = cur + N;

    // CSR build
    int zn = 3 * N + 1; // cnt, offs, cur are contiguous
    k_zero_i32<<<(zn + 255) / 256, 256, 0, stream>>>(cnt, zn);
    k_count<<<(E + 255) / 256, 256, 0, stream>>>(e_dst, cnt, E);
    k_scan<<<1, 1024, 0, stream>>>(cnt, offs, N);
    k_fill<<<(E + 255) / 256, 256, 0, stream>>>(e_src, e_dst, offs, cur, csr, E);

    // layer 1
    gat_gemm1<<<dim3(N / 16, HEADS), 128, 0, stream>>>(x, W1, a_src1, a_dst1, h1, als1, ald1);
    gat_agg1<<<N, 128, 0, stream>>>(h1, als1, ald1, offs, csr, b1, gam, bet, bmean, bvar, x1);

    // layer 2
    gat_gemm2<<<N / 16, 64, 0, stream>>>(x1, W2, a_src2, a_dst2, h2, als2, ald2);
    gat_agg2<<<N, 32, 0, stream>>>(h2, als2, ald2, offs, csr, b2, out);
}